// Mamba_33371895890357
// MI455X (gfx1250) — compile-verified
//
#include <hip/hip_runtime.h>
#include <hip/hip_bf16.h>
#include <math.h>

// Problem constants
#define L_TOK   4096      // H*W
#define D_IN    192       // D_INNER
#define N_ST    16        // D_STATE
#define C_CH    96
#define NCHUNK  64
#define CHLEN   64        // NCHUNK*CHLEN == L_TOK

typedef __attribute__((ext_vector_type(2))) float v2f;
typedef __attribute__((ext_vector_type(8))) float v8f;

__device__ __forceinline__ v8f wmma4(v2f a, v2f b, v8f c) {
  // V_WMMA_F32_16X16X4_F32 : D = A(16x4) * B(4x16) + C(16x16), full f32
  return __builtin_amdgcn_wmma_f32_16x16x4_f32(false, a, false, b, (short)0, c,
                                               false, false);
}

// scan-step l  ->  spatial pixel (row-major h*W+w) for direction k
__device__ __forceinline__ int dir_pixel(int k, int l) {
  int ll = (k & 2) ? (L_TOK - 1 - l) : l;
  if (k & 1) return ((ll & 63) << 6) | (ll >> 6); // H<->W transpose (involution)
  return ll;
}

// ---------------------------------------------------------------------------
// 0) weight transpose for the conv implicit GEMM:
//    cw[cout][ci][kh][kw]  ->  cwt[cout][tap][ci]   (K-pairs contiguous so the
//    WMMA B-fragment pair loads merge into global_load_b64)
// ---------------------------------------------------------------------------
__global__ void k_wtr(const float* __restrict__ cw, float* __restrict__ cwt) {
  int i = blockIdx.x * blockDim.x + threadIdx.x;      // over 96*1728
  if (i >= C_CH * 1728) return;
  int cout = i / 1728, r = i % 1728;
  int tap = r / 192, ci = r % 192;
  cwt[i] = cw[(size_t)cout * 1728 + ci * 9 + tap];
}

// ---------------------------------------------------------------------------
// 1) conv3x3 (concat(rgb,t): 192 -> 96) as implicit-GEMM WMMA + BN+ReLU+LN.
//    Block = 16 pixels x 96 cout, 192 threads = 6 waves, one 16-wide cout
//    tile per wave. A fragments come from the LDS patch, K = 9 taps * 192 cin.
// ---------------------------------------------------------------------------
__global__ void __launch_bounds__(192)
k_convw(const float* __restrict__ rgb, const float* __restrict__ tin,
        const float* __restrict__ cwt, const float* __restrict__ cb,
        const float* __restrict__ bng, const float* __restrict__ bnb,
        const float* __restrict__ bnm, const float* __restrict__ bnv,
        const float* __restrict__ lng, const float* __restrict__ lnb,
        float* __restrict__ xln) {
  __shared__ float sm[3 * 18 * 192];   // input patch, 41.5 KB of 320 KB WGP LDS
  const int tid  = threadIdx.x;
  const int lane = tid & 31;
  const int n0   = (tid >> 5) << 4;    // wave -> cout tile base
  const int w0   = blockIdx.x * 16;
  const int h    = blockIdx.y;
  const int b    = blockIdx.z;

  for (int i = tid; i < 3 * 18 * 192; i += 192) {
    int ci = i % 192;
    int cc = (i / 192) % 18;
    int r  = i / (192 * 18);
    int gh = h + r - 1, gw = w0 + cc - 1;
    float v = 0.f;
    if (gh >= 0 && gh < 64 && gw >= 0 && gw < 64)
      v = (ci < 96) ? rgb[((b * 96 + ci) * 64 + gh) * 64 + gw]
                    : tin[((b * 96 + ci - 96) * 64 + gh) * 64 + gw];
    sm[(r * 18 + cc) * 192 + ci] = v;
  }
  __syncthreads();

  const int px    = lane & 15;          // pixel row of A fragment
  const int cq    = n0 + (lane & 15);   // cout column of B fragment
  const int khalf = (lane >> 4) << 1;
  v8f acc = {0.f, 0.f, 0.f, 0.f, 0.f, 0.f, 0.f, 0.f};
#pragma unroll
  for (int tap = 0; tap < 9; tap++) {
    const int kh = tap / 3, kw = tap % 3;
    const float* smrow = sm + (kh * 18 + px + kw) * 192;           // A(px, k)
    const float* wrow  = cwt + (size_t)cq * 1728 + tap * 192;      // B(k, cout)
#pragma unroll 4
    for (int ci = 0; ci < 192; ci += 4) {
      int ka = ci + khalf;
      v2f a, bv;
      a.x  = smrow[ka];
      a.y  = smrow[ka + 1];
      bv.x = wrow[ka];          // contiguous K-pair -> global_load_b64
      bv.y = wrow[ka + 1];
      acc = wmma4(a, bv, acc);
    }
  }
  __syncthreads();                      // patch dead, reuse LDS for LN

  // epilogue: bias + BN + ReLU into LDS vals[px][cout]
  float rsb = rsqrtf(bnv[cq] + 1e-5f);
  float cbv = cb[cq], mv = bnm[cq], gv = bng[cq], btv = bnb[cq];
  int pxb = (lane >> 4) << 3;
#pragma unroll
  for (int j = 0; j < 8; j++) {
    float x = acc[j] + cbv;
    x = (x - mv) * rsb * gv + btv;
    sm[(pxb + j) * 96 + cq] = fmaxf(x, 0.f);
  }
  __syncthreads();
  if (tid < 16) {                       // per-pixel LN stats over 96 channels
    float s1 = 0.f, s2 = 0.f;
    for (int j = 0; j < 96; j++) { float v = sm[tid * 96 + j]; s1 += v; s2 += v * v; }
    float mu  = s1 / 96.f;
    float var = s2 / 96.f - mu * mu;
    sm[1600 + tid] = mu;
    sm[1632 + tid] = rsqrtf(var + 1e-5f);
  }
  __syncthreads();
  for (int i = tid; i < 16 * 96; i += 192) {
    int ppx = i / 96, c = i % 96;
    float o = (sm[i] - sm[1600 + ppx]) * sm[1632 + ppx] * lng[c] + lnb[c];
    xln[((size_t)b * L_TOK + h * 64 + w0 + ppx) * 96 + c] = o;
  }
}

// ---------------------------------------------------------------------------
// 2) in_proj : xz[b,p,e] = sum_c xln[b,p,c] * W[e,c]   (M=8192,N=384,K=96)
// ---------------------------------------------------------------------------
__global__ void __launch_bounds__(32) k_inproj(const float* __restrict__ xln,
                                               const float* __restrict__ w,
                                               float* __restrict__ xz) {
  const int m0 = blockIdx.x * 16, n0 = blockIdx.y * 16;
  const int lane  = threadIdx.x;
  const int mrow  = m0 + (lane & 15);
  const int ncol  = n0 + (lane & 15);
  const int khalf = (lane >> 4) << 1;
  v8f acc = {0.f, 0.f, 0.f, 0.f, 0.f, 0.f, 0.f, 0.f};
#pragma unroll 6
  for (int k0 = 0; k0 < 96; k0 += 4) {
    int ka = k0 + khalf;
    v2f a, bv;
    a.x  = xln[(size_t)mrow * 96 + ka];
    a.y  = xln[(size_t)mrow * 96 + ka + 1];
    bv.x = w[(size_t)ncol * 96 + ka];       // B(k=c, n=e) = W[e*96+c]
    bv.y = w[(size_t)ncol * 96 + ka + 1];
    acc = wmma4(a, bv, acc);
  }
  int rbase = m0 + ((lane >> 4) << 3);
  int col   = n0 + (lane & 15);
#pragma unroll
  for (int j = 0; j < 8; j++)
    xz[(size_t)(rbase + j) * 384 + col] = acc[j];
}

// ---------------------------------------------------------------------------
// 3) depthwise conv3x3 + bias + SiLU  : xz[:, :192] -> xca (B,L,192)
// ---------------------------------------------------------------------------
__global__ void k_dwconv(const float* __restrict__ xz, const float* __restrict__ dww,
                         const float* __restrict__ dwb, float* __restrict__ xca) {
  int idx = blockIdx.x * blockDim.x + threadIdx.x;
  if (idx >= 2 * L_TOK * D_IN) return;
  int d = idx % D_IN;
  int p = (idx / D_IN) & (L_TOK - 1);
  int b = idx / (D_IN * L_TOK);
  int h = p >> 6, w = p & 63;
  float s = 0.f;
#pragma unroll
  for (int kh = 0; kh < 3; kh++) {
    int gh = h + kh - 1;
    if (gh < 0 || gh >= 64) continue;
#pragma unroll
    for (int kw = 0; kw < 3; kw++) {
      int gw = w + kw - 1;
      if (gw < 0 || gw >= 64) continue;
      s += dww[d * 9 + kh * 3 + kw] *
           xz[((size_t)b * L_TOK + gh * 64 + gw) * 384 + d];
    }
  }
  s += dwb[d];
  xca[(size_t)idx] = s / (1.f + __expf(-s));   // SiLU
}

// ---------------------------------------------------------------------------
// 4) x_proj per (b,dir): x_dbl[l,c] = sum_d xs[d,l]*xpw[dir,c,d]
//    M=4096, N=38 padded to 48, K=192. A rows remapped by scan direction.
// ---------------------------------------------------------------------------
__global__ void __launch_bounds__(32) k_xproj(const float* __restrict__ xca,
                                              const float* __restrict__ xpw,
                                              float* __restrict__ dtr,
                                              float* __restrict__ bsb,
                                              float* __restrict__ csb) {
  const int bk = blockIdx.z, b = bk >> 2, kd = bk & 3;
  const int m0 = blockIdx.x * 16, n0 = blockIdx.y * 16;
  const int lane = threadIdx.x;
  const int lrow = m0 + (lane & 15);
  const int pix  = dir_pixel(kd, lrow);
  const float* arow = xca + ((size_t)b * L_TOK + pix) * D_IN;
  const int nc    = n0 + (lane & 15);
  const int khalf = (lane >> 4) << 1;
  v8f acc = {0.f, 0.f, 0.f, 0.f, 0.f, 0.f, 0.f, 0.f};
#pragma unroll 4
  for (int k0 = 0; k0 < D_IN; k0 += 4) {
    int ka = k0 + khalf;
    v2f a, bv;
    a.x  = arow[ka];
    a.y  = arow[ka + 1];
    bv.x = (nc < 38) ? xpw[(size_t)(kd * 38 + nc) * D_IN + ka] : 0.f;
    bv.y = (nc < 38) ? xpw[(size_t)(kd * 38 + nc) * D_IN + ka + 1] : 0.f;
    acc = wmma4(a, bv, acc);
  }
  int col   = n0 + (lane & 15);
  int rbase = m0 + ((lane >> 4) << 3);
#pragma unroll
  for (int j = 0; j < 8; j++) {
    int l = rbase + j;
    float v = acc[j];
    size_t base = (size_t)bk * L_TOK + l;
    if (col < 6)        dtr[base * 8 + col] = v;          // dt_rank part (stride 8)
    else if (col < 22)  bsb[base * 16 + (col - 6)] = v;   // Bs
    else if (col < 38)  csb[base * 16 + (col - 22)] = v;  // Cs
  }
}

// ---------------------------------------------------------------------------
// 5) dt GEMM per (b,dir): delta_pre[l,d] = sum_r dtr[l,r]*dt_w[dir,d,r]
//    M=4096, N=192, K=6 padded to 8 (masked loaders)
// ---------------------------------------------------------------------------
__global__ void __launch_bounds__(32) k_dt(const float* __restrict__ dtr,
                                           const float* __restrict__ dtw,
                                           float* __restrict__ dpre) {
  const int bk = blockIdx.z, kd = bk & 3;
  const int m0 = blockIdx.x * 16, n0 = blockIdx.y * 16;
  const int lane  = threadIdx.x;
  const int lrow  = m0 + (lane & 15);
  const int nc    = n0 + (lane & 15);
  const int khalf = (lane >> 4) << 1;
  v8f acc = {0.f, 0.f, 0.f, 0.f, 0.f, 0.f, 0.f, 0.f};
#pragma unroll
  for (int k0 = 0; k0 < 8; k0 += 4) {
    int ka = k0 + khalf;
    v2f a, bv;
    a.x  = (ka     < 6) ? dtr[((size_t)bk * L_TOK + lrow) * 8 + ka]     : 0.f;
    a.y  = (ka + 1 < 6) ? dtr[((size_t)bk * L_TOK + lrow) * 8 + ka + 1] : 0.f;
    bv.x = (ka     < 6) ? dtw[(size_t)(kd * D_IN + nc) * 6 + ka]     : 0.f;
    bv.y = (ka + 1 < 6) ? dtw[(size_t)(kd * D_IN + nc) * 6 + ka + 1] : 0.f;
    acc = wmma4(a, bv, acc);
  }
  int col   = n0 + (lane & 15);
  int rbase = m0 + ((lane >> 4) << 3);
#pragma unroll
  for (int j = 0; j < 8; j++)
    dpre[((size_t)bk * L_TOK + rbase + j) * D_IN + col] = acc[j];
}

// ---------------------------------------------------------------------------
// 6) Chunked parallel scan over the associative recurrence h <- e*h + u.
//    Phase A: per chunk of 64 steps, compute decay product P = prod(e) and
//             local result hloc (state starting from 0).
//    Phase B: sequential carry combine over the 64 chunks (one thread per
//             (b,dir,d,n) = 24576 threads, 64 steps).
//    Phase C: replay each chunk from its carry, emit y scattered to final
//             pixel positions (folds the 4-direction recombination).
//    Critical path 4096 -> ~192 steps (3 x 64); recompute is cheap exp/FMA work.
// ---------------------------------------------------------------------------
__global__ void __launch_bounds__(192)
k_scanA(const float* __restrict__ dpre, const float* __restrict__ xca,
        const float* __restrict__ bsb,  const float* __restrict__ dtb,
        const float* __restrict__ alogs,
        float* __restrict__ pbuf, float* __restrict__ hbuf) {
  const int chunk = blockIdx.x, bk = blockIdx.y, b = bk >> 2, kd = bk & 3;
  const int d = threadIdx.x;
  float A[N_ST], P[N_ST], hv[N_ST];
#pragma unroll
  for (int n = 0; n < N_ST; n++) {
    A[n] = -__expf(alogs[(size_t)(kd * D_IN + d) * N_ST + n]);
    P[n] = 1.f; hv[n] = 0.f;
  }
  const float dtbv = dtb[kd * D_IN + d];
  const size_t baseL = (size_t)bk * L_TOK;
  for (int i = 0; i < CHLEN; i++) {
    int l = chunk * CHLEN + i;
    int p = dir_pixel(kd, l);
    __builtin_prefetch(&dpre[(baseL + l + 16) * D_IN + d], 0, 0);
    float xv = xca[((size_t)b * L_TOK + p) * D_IN + d];
    float dl = dpre[(baseL + l) * D_IN + d] + dtbv;
    dl = (dl > 20.f) ? dl : __logf(1.f + __expf(dl));   // softplus
    float du = dl * xv;
    const float* Bl = bsb + (baseL + l) * N_ST;         // wave-broadcast loads
#pragma unroll
    for (int n = 0; n < N_ST; n++) {
      float e = __expf(dl * A[n]);
      P[n] *= e;
      hv[n] = hv[n] * e + du * Bl[n];
    }
  }
  size_t ob = ((size_t)(bk * NCHUNK + chunk) * D_IN + d) * N_ST;
#pragma unroll
  for (int n = 0; n < N_ST; n++) { pbuf[ob + n] = P[n]; hbuf[ob + n] = hv[n]; }
}

__global__ void k_scanB(const float* __restrict__ pbuf,
                        const float* __restrict__ hbuf,
                        float* __restrict__ cbuf) {
  int t = blockIdx.x * blockDim.x + threadIdx.x;
  if (t >= 8 * D_IN * N_ST) return;
  int bk = t / (D_IN * N_ST);
  int r  = t % (D_IN * N_ST);
  int d  = r >> 4, n = r & 15;
  float carry = 0.f;
  for (int c = 0; c < NCHUNK; c++) {
    size_t idx = ((size_t)(bk * NCHUNK + c) * D_IN + d) * N_ST + n;
    cbuf[idx] = carry;                       // state entering chunk c
    carry = pbuf[idx] * carry + hbuf[idx];
  }
}

__global__ void __launch_bounds__(192)
k_scanC(const float* __restrict__ dpre, const float* __restrict__ xca,
        const float* __restrict__ bsb,  const float* __restrict__ csb,
        const float* __restrict__ dtb,  const float* __restrict__ alogs,
        const float* __restrict__ dsv,  const float* __restrict__ cbuf,
        float* __restrict__ ys) {
  const int chunk = blockIdx.x, bk = blockIdx.y, b = bk >> 2, kd = bk & 3;
  const int d = threadIdx.x;
  float A[N_ST], hv[N_ST];
  size_t ob = ((size_t)(bk * NCHUNK + chunk) * D_IN + d) * N_ST;
#pragma unroll
  for (int n = 0; n < N_ST; n++) {
    A[n]  = -__expf(alogs[(size_t)(kd * D_IN + d) * N_ST + n]);
    hv[n] = cbuf[ob + n];
  }
  const float dtbv = dtb[kd * D_IN + d];
  const float Dv   = dsv[kd * D_IN + d];
  const size_t baseL = (size_t)bk * L_TOK;
  for (int i = 0; i < CHLEN; i++) {
    int l = chunk * CHLEN + i;
    int p = dir_pixel(kd, l);
    __builtin_prefetch(&dpre[(baseL + l + 16) * D_IN + d], 0, 0);
    float xv = xca[((size_t)b * L_TOK + p) * D_IN + d];
    float dl = dpre[(baseL + l) * D_IN + d] + dtbv;
    dl = (dl > 20.f) ? dl : __logf(1.f + __expf(dl));
    float du = dl * xv;
    const float* Bl = bsb + (baseL + l) * N_ST;
    const float* Cl = csb + (baseL + l) * N_ST;
    float y = 0.f;
#pragma unroll
    for (int n = 0; n < N_ST; n++) {
      hv[n] = hv[n] * __expf(dl * A[n]) + du * Bl[n];
      y += hv[n] * Cl[n];
    }
    ys[(baseL + p) * D_IN + d] = y + xv * Dv;           // scatter to pixel p
  }
}

// ---------------------------------------------------------------------------
// 7) sum 4 directions + out-LayerNorm + SiLU(z) gate : one block per token
// ---------------------------------------------------------------------------
__global__ void __launch_bounds__(192) k_gate(const float* __restrict__ ys,
                                              const float* __restrict__ xz,
                                              const float* __restrict__ g,
                                              const float* __restrict__ bt,
                                              float* __restrict__ yg) {
  __shared__ float sm[384];
  const int d = threadIdx.x;
  const int p = blockIdx.x & (L_TOK - 1);
  const int b = blockIdx.x >> 12;
  const size_t bl = (size_t)b * L_TOK + p;
  float v = 0.f;
#pragma unroll
  for (int k = 0; k < 4; k++)
    v += ys[((size_t)(b * 4 + k) * L_TOK + p) * D_IN + d];
  sm[d] = v; sm[192 + d] = v * v;
  __syncthreads();
  for (int s = 96; s >= 3; s >>= 1) {
    if (d < s) { sm[d] += sm[d + s]; sm[192 + d] += sm[192 + d + s]; }
    __syncthreads();
  }
  if (d == 0) {
    float s1 = sm[0] + sm[1] + sm[2];
    float s2 = sm[192] + sm[193] + sm[194];
    float mu  = s1 / 192.f;
    float var = s2 / 192.f - mu * mu;
    sm[0] = mu; sm[1] = rsqrtf(var + 1e-5f);
  }
  __syncthreads();
  float mu = sm[0], rs = sm[1];
  float z  = xz[bl * 384 + 192 + d];
  float ln = (v - mu) * rs * g[d] + bt[d];
  yg[bl * D_IN + d] = ln * (z / (1.f + __expf(-z)));
}

// ---------------------------------------------------------------------------
// 8) out_proj: out[b,c,p] = sum_d yg[b,p,d]*W[c,d]  (M=8192,N=96,K=192)
//    epilogue stores transposed into NCHW output
// ---------------------------------------------------------------------------
__global__ void __launch_bounds__(32) k_outproj(const float* __restrict__ yg,
                                                const float* __restrict__ w,
                                                float* __restrict__ out) {
  const int m0 = blockIdx.x * 16, n0 = blockIdx.y * 16;
  const int lane  = threadIdx.x;
  const int mrow  = m0 + (lane & 15);
  const int ncol  = n0 + (lane & 15);
  const int khalf = (lane >> 4) << 1;
  v8f acc = {0.f, 0.f, 0.f, 0.f, 0.f, 0.f, 0.f, 0.f};
#pragma unroll 4
  for (int k0 = 0; k0 < D_IN; k0 += 4) {
    int ka = k0 + khalf;
    v2f a, bv;
    a.x  = yg[(size_t)mrow * D_IN + ka];
    a.y  = yg[(size_t)mrow * D_IN + ka + 1];
    bv.x = w[(size_t)ncol * D_IN + ka];     // B(k=d, n=c) = W[c*192+d]
    bv.y = w[(size_t)ncol * D_IN + ka + 1];
    acc = wmma4(a, bv, acc);
  }
  int rbase = m0 + ((lane >> 4) << 3);
  int col   = n0 + (lane & 15);
#pragma unroll
  for (int j = 0; j < 8; j++) {
    int m = rbase + j;
    int b = m >> 12, p = m & (L_TOK - 1);
    out[((size_t)b * C_CH + col) * L_TOK + p] = acc[j];
  }
}

// ---------------------------------------------------------------------------
extern "C" void kernel_launch(void* const* d_in, const int* in_sizes, int n_in,
                              void* d_out, int out_size, void* d_ws, size_t ws_size,
                              hipStream_t stream) {
  (void)in_sizes; (void)n_in; (void)out_size; (void)ws_size;
  const float* rgb   = (const float*)d_in[0];
  const float* tin   = (const float*)d_in[1];
  const float* cw    = (const float*)d_in[2];
  const float* cb    = (const float*)d_in[3];
  const float* bng   = (const float*)d_in[4];
  const float* bnb   = (const float*)d_in[5];
  const float* bnm   = (const float*)d_in[6];
  const float* bnv   = (const float*)d_in[7];
  const float* lng   = (const float*)d_in[8];
  const float* lnb   = (const float*)d_in[9];
  const float* ipw   = (const float*)d_in[10];
  const float* dww   = (const float*)d_in[11];
  const float* dwb   = (const float*)d_in[12];
  const float* xpw   = (const float*)d_in[13];
  const float* dtw   = (const float*)d_in[14];
  const float* dtb   = (const float*)d_in[15];
  const float* alogs = (const float*)d_in[16];
  const float* dsv   = (const float*)d_in[17];
  const float* ong   = (const float*)d_in[18];
  const float* onb   = (const float*)d_in[19];
  const float* opw   = (const float*)d_in[20];
  float* out = (float*)d_out;

  // workspace layout (floats): ~25.9M floats = 104 MB (fits MI455X 192MB L2)
  float* ws   = (float*)d_ws;
  size_t o = 0;
  float* xln  = ws + o; o += (size_t)2 * L_TOK * 96;            // 786432
  float* xz   = ws + o; o += (size_t)2 * L_TOK * 384;           // 3145728
  float* xca  = ws + o; o += (size_t)2 * L_TOK * D_IN;          // 1572864
  float* dtr  = ws + o; o += (size_t)8 * L_TOK * 8;             // 262144
  float* bsb  = ws + o; o += (size_t)8 * L_TOK * 16;            // 524288
  float* csb  = ws + o; o += (size_t)8 * L_TOK * 16;            // 524288
  float* dpre = ws + o; o += (size_t)8 * L_TOK * D_IN;          // 6291456
  float* ysb  = ws + o; o += (size_t)8 * L_TOK * D_IN;          // 6291456
  float* yg   = ws + o; o += (size_t)2 * L_TOK * D_IN;          // 1572864
  float* pbuf = ws + o; o += (size_t)8 * NCHUNK * D_IN * N_ST;  // 1572864
  float* hbuf = ws + o; o += (size_t)8 * NCHUNK * D_IN * N_ST;  // 1572864
  float* cbuf = ws + o; o += (size_t)8 * NCHUNK * D_IN * N_ST;  // 1572864
  float* cwt  = ws + o; o += (size_t)C_CH * 1728;               // 165888

  k_wtr<<<dim3((C_CH * 1728 + 255) / 256), 256, 0, stream>>>(cw, cwt);
  k_convw<<<dim3(4, 64, 2), 192, 0, stream>>>(rgb, tin, cwt, cb, bng, bnb, bnm,
                                              bnv, lng, lnb, xln);
  k_inproj<<<dim3(512, 24), 32, 0, stream>>>(xln, ipw, xz);
  k_dwconv<<<dim3((2 * L_TOK * D_IN + 255) / 256), 256, 0, stream>>>(xz, dww, dwb, xca);
  k_xproj<<<dim3(256, 3, 8), 32, 0, stream>>>(xca, xpw, dtr, bsb, csb);
  k_dt<<<dim3(256, 12, 8), 32, 0, stream>>>(dtr, dtw, dpre);
  k_scanA<<<dim3(NCHUNK, 8), 192, 0, stream>>>(dpre, xca, bsb, dtb, alogs, pbuf, hbuf);
  k_scanB<<<dim3((8 * D_IN * N_ST + 255) / 256), 256, 0, stream>>>(pbuf, hbuf, cbuf);
  k_scanC<<<dim3(NCHUNK, 8), 192, 0, stream>>>(dpre, xca, bsb, csb, dtb, alogs,
                                               dsv, cbuf, ysb);
  k_gate<<<dim3(2 * L_TOK), 192, 0, stream>>>(ysb, xz, ong, onb, yg);
  k_outproj<<<dim3(512, 6), 32, 0, stream>>>(yg, opw, out);
}